// node_update_mix_70961449664975
// MI455X (gfx1250) — compile-verified
//
#include <hip/hip_runtime.h>
#include <hip/hip_bf16.h>
#include <math.h>

// ---------------------------------------------------------------------------
// node_update_mix for MI455X (gfx1250, wave32, WMMA)
//   agg = segment_sum(edge_feats, edge_index[1], N)        (scatter, BW-bound)
//   s   = agg.sum(axis=1)
//   h   = softplus(s@W1+b1) - log2 ; h = h@W2 + b2
//   out0 = node_sca + h ; out1 = node_vec + agg[:,1:,:]
// GEMMs use V_WMMA_F32_16X16X4_F32 (exact fp32; GEMM cost is negligible vs
// the 819MB edge-read floor, so no precision is sacrificed for speed).
// ---------------------------------------------------------------------------

typedef __attribute__((ext_vector_type(2))) float v2f;
typedef __attribute__((ext_vector_type(8))) float v8f;

#define H_DIM 128
#define F_DIM 128
#define K_DIM 4
#define ROW_STRIDE 512        // K_DIM * F_DIM
#define LDSP 132              // padded LDS row stride (132 % 64 = 4 -> conflict-free)

// ---------------- kernel 1: zero the accumulator ---------------------------
__global__ __launch_bounds__(256) void zero_f4_kernel(float4* __restrict__ p, size_t n4) {
  size_t t = (size_t)blockIdx.x * blockDim.x + threadIdx.x;
  if (t < n4) p[t] = make_float4(0.f, 0.f, 0.f, 0.f);
}

// ---------------- kernel 2: edge scatter-add (segment_sum) -----------------
// One float4 of edge_feats per thread; 4 f32 atomics that resolve in L2
// (51MB accumulator is L2-resident on the 192MB L2).
__global__ __launch_bounds__(256) void edge_scatter_kernel(
    const float* __restrict__ edge_feats, const int* __restrict__ dst_idx,
    float* __restrict__ agg, size_t n4) {
  size_t t = (size_t)blockIdx.x * blockDim.x + threadIdx.x;
  if (t >= n4) return;
  size_t base = t * 4;                 // flat element index into (E,4,128)
  size_t e    = base >> 9;             // / 512
  int    rem  = (int)(base & 511);
  int    node = dst_idx[e];
  float4 v = ((const float4*)edge_feats)[t];
  float* d = agg + (size_t)node * ROW_STRIDE + rem;
  unsafeAtomicAdd(d + 0, v.x);
  unsafeAtomicAdd(d + 1, v.y);
  unsafeAtomicAdd(d + 2, v.z);
  unsafeAtomicAdd(d + 3, v.w);
}

// ---------------- kernel 3: fused 2-layer MLP via WMMA ---------------------
// One block = 16 node rows. 8 waves; wave w owns output columns [16w,16w+16).
// GEMM1 (s@W1) -> shifted softplus -> LDS -> GEMM2 (h@W2) -> +node_sca.
__global__ __launch_bounds__(256) void fused_mlp_kernel(
    const float* __restrict__ agg, const float* __restrict__ node_sca,
    const float* __restrict__ W1, const float* __restrict__ b1,
    const float* __restrict__ W2, const float* __restrict__ b2,
    float* __restrict__ out0, int N) {
  __shared__ float sS[16 * LDSP];   // s tile  (16 x 128, padded)
  __shared__ float sH[16 * LDSP];   // h tile  (16 x 128, padded)

  const int node0 = blockIdx.x * 16;
  const int tid   = threadIdx.x;

  // Stage s = sum over K of agg for 16 rows (2048 elems / 256 threads).
  for (int idx = tid; idx < 16 * H_DIM; idx += 256) {
    int m = idx >> 7, k = idx & 127;
    int node = node0 + m;
    float v = 0.f;
    if (node < N) {
      const float* p = agg + (size_t)node * ROW_STRIDE + k;
      v = p[0] + p[128] + p[256] + p[384];
    }
    sS[m * LDSP + k] = v;
  }
  __syncthreads();

  const int lane  = tid & 31;
  const int wave  = tid >> 5;
  const int col   = wave * 16 + (lane & 15);  // output column (N index)
  const int m     = lane & 15;                // A-row this lane feeds
  const int koff  = (lane >> 4) << 1;         // upper half-wave -> K+2
  const int mbase = (lane >> 4) << 3;         // C/D rows: v + 8*(lane>=16)

  // ---- GEMM1: acc = s @ W1 + b1 (bias folded into C) ----
  v8f acc;
  {
    float bias = b1[col];
    #pragma unroll
    for (int v = 0; v < 8; ++v) acc[v] = bias;
  }
  #pragma unroll
  for (int k0 = 0; k0 < H_DIM; k0 += 4) {
    v2f a, b;
    a[0] = sS[m * LDSP + k0 + koff];
    a[1] = sS[m * LDSP + k0 + koff + 1];
    b[0] = W1[(size_t)(k0 + koff) * H_DIM + col];
    b[1] = W1[(size_t)(k0 + koff + 1) * H_DIM + col];
    acc = __builtin_amdgcn_wmma_f32_16x16x4_f32(
        false, a, false, b, (short)0, acc, false, false);
  }

  // shifted softplus: max(x,0) + log1p(exp(-|x|)) - log(2)   (stable)
  #pragma unroll
  for (int v = 0; v < 8; ++v) {
    float x  = acc[v];
    float sp = fmaxf(x, 0.f) + log1pf(__expf(-fabsf(x)));
    sH[(mbase + v) * LDSP + col] = sp - 0.69314718055994531f;
  }
  __syncthreads();

  // ---- GEMM2: acc2 = h @ W2 + b2 ----
  v8f acc2;
  {
    float bias = b2[col];
    #pragma unroll
    for (int v = 0; v < 8; ++v) acc2[v] = bias;
  }
  #pragma unroll
  for (int k0 = 0; k0 < H_DIM; k0 += 4) {
    v2f a, b;
    a[0] = sH[m * LDSP + k0 + koff];
    a[1] = sH[m * LDSP + k0 + koff + 1];
    b[0] = W2[(size_t)(k0 + koff) * H_DIM + col];
    b[1] = W2[(size_t)(k0 + koff + 1) * H_DIM + col];
    acc2 = __builtin_amdgcn_wmma_f32_16x16x4_f32(
        false, a, false, b, (short)0, acc2, false, false);
  }

  // out0 = node_sca + h2
  #pragma unroll
  for (int v = 0; v < 8; ++v) {
    int node = node0 + mbase + v;
    if (node < N) {
      size_t o = (size_t)node * H_DIM + col;
      out0[o] = node_sca[o] + acc2[v];
    }
  }
}

// ---------------- kernel 4: out1 = node_vec + agg[:,1:,:] ------------------
__global__ __launch_bounds__(256) void vec_add_kernel(
    const float* __restrict__ node_vec, const float* __restrict__ agg,
    float* __restrict__ out1, size_t n4) {
  size_t t = (size_t)blockIdx.x * blockDim.x + threadIdx.x;
  if (t >= n4) return;
  size_t base = t * 4;                 // flat into (N,3,128); 384 % 4 == 0
  size_t n    = base / 384;
  int    rem  = (int)(base % 384);
  float4 a = *(const float4*)(node_vec + base);
  float4 g = *(const float4*)(agg + n * ROW_STRIDE + F_DIM + rem);
  *(float4*)(out1 + base) =
      make_float4(a.x + g.x, a.y + g.y, a.z + g.z, a.w + g.w);
}

// ---------------------------------------------------------------------------
extern "C" void kernel_launch(void* const* d_in, const int* in_sizes, int n_in,
                              void* d_out, int out_size, void* d_ws, size_t ws_size,
                              hipStream_t stream) {
  const float* node_sca   = (const float*)d_in[0];
  const float* node_vec   = (const float*)d_in[1];
  const float* edge_feats = (const float*)d_in[2];
  const int*   edge_index = (const int*)d_in[3];   // (2,E); row 1 = dst
  const float* W1 = (const float*)d_in[4];
  const float* b1 = (const float*)d_in[5];
  const float* W2 = (const float*)d_in[6];
  const float* b2 = (const float*)d_in[7];

  const int N = in_sizes[0] / H_DIM;
  const int E = in_sizes[3] / 2;

  float* agg  = (float*)d_ws;                       // (N,4,128) accumulator
  float* out0 = (float*)d_out;                      // (N,128)
  float* out1 = out0 + (size_t)N * H_DIM;           // (N,3,128)

  // 1) zero accumulator (atomics below accumulate into it every call)
  size_t z4 = (size_t)N * ROW_STRIDE / 4;
  zero_f4_kernel<<<(unsigned)((z4 + 255) / 256), 256, 0, stream>>>((float4*)agg, z4);

  // 2) segment-sum scatter (dominant: 819MB edge read @ 23.3TB/s)
  size_t s4 = (size_t)E * ROW_STRIDE / 4;
  edge_scatter_kernel<<<(unsigned)((s4 + 255) / 256), 256, 0, stream>>>(
      edge_feats, edge_index + E, agg, s4);

  // 3) fused MLP + out0
  fused_mlp_kernel<<<(N + 15) / 16, 256, 0, stream>>>(
      agg, node_sca, W1, b1, W2, b2, out0, N);

  // 4) out1 = node_vec + agg[:,1:,:]
  size_t v4 = (size_t)N * 3 * F_DIM / 4;
  vec_add_kernel<<<(unsigned)((v4 + 255) / 256), 256, 0, stream>>>(
      node_vec, agg, out1, v4);
}